// PatchICL_56427280335133
// MI455X (gfx1250) — compile-verified
//
#include <hip/hip_runtime.h>
#include <stdint.h>

// ---------------------------------------------------------------------------
// PatchICL for MI455X (gfx1250, wave32, WMMA + async LDS staging).
//
// Roofline: ~110 MB total traffic @ 23.3 TB/s -> ~5us floor; FLOPs negligible
// => memory-bound. The feature resampler computes  out = Wy * F * Wx^T  per
// 16-channel slab with V_WMMA_F32_16X16X4_F32 (exact f32, matches reference).
// Data path per wave-slab:
//   global --(global_load_async_to_lds_b128, ASYNCcnt)--> LDS window W
//   W --(ds_load_2addr)--> WMMA(x-interp) --> LDS tile T (layout transpose)
//   T --(ds_load_2addr)--> WMMA(y-interp) --> coalesced global stores
// Grid is split so every wave owns one (patch, slab): 4096 independent waves.
// ---------------------------------------------------------------------------

typedef __attribute__((ext_vector_type(2))) float v2f;
typedef __attribute__((ext_vector_type(4))) float v4f;
typedef __attribute__((ext_vector_type(8))) float v8f;

constexpr int B_ = 4, K_ = 32, D_ = 512, N_ = 4096;
constexpr int RES = 512;
constexpr int CONF_N  = B_ * K_ * 64 * 64;  // 524288
constexpr int MASKP_N = B_ * K_ * 16 * 16;  // 32768

__device__ __forceinline__ float clampf(float v, float lo, float hi) {
  return fminf(fmaxf(v, lo), hi);
}

// ------------------------------- confidence --------------------------------
__global__ void conf_kernel(const float* __restrict__ logits,
                            float* __restrict__ out) {
  int i = blockIdx.x * blockDim.x + threadIdx.x;  // over float4 groups
  v4f z4 = ((const v4f*)logits)[i];
  v4f r;
#pragma unroll
  for (int j = 0; j < 4; j++) {
    float z = z4[j] * (1.0f / 1.5f);
    float p = 1.0f / (1.0f + __expf(-z));
    p = clampf(p, 1e-6f, 1.0f - 1e-6f);
    float ent = -(p * __logf(p) + (1.0f - p) * __logf(1.0f - p));
    r[j] = 1.0f - ent * 1.4426950408889634f;  // 1/ln(2)
  }
  ((v4f*)out)[i] = r;
}

// ------------------------------ mask patches -------------------------------
__global__ void mask_kernel(const float* __restrict__ M,
                            const int* __restrict__ coords,
                            float* __restrict__ out) {
  const int bk = blockIdx.x;
  const int b = bk >> 5;
  const int t = threadIdx.x;
  const int ty = t >> 4, tx = t & 15;
  const float ch = (float)coords[bk * 2 + 0];
  const float cw = (float)coords[bk * 2 + 1];
  const float cy = (ch + 32.0f) * (1.0f / 256.0f) - 1.0f;
  const float cx = (cw + 32.0f) * (1.0f / 256.0f) - 1.0f;
  const float oy = -0.125f + (float)ty * (0.25f / 15.0f);
  const float ox = -0.125f + (float)tx * (0.25f / 15.0f);
  float y = clampf((cy + oy + 1.0f) * 256.0f - 0.5f, 0.0f, 511.0f);
  float x = clampf((cx + ox + 1.0f) * 256.0f - 0.5f, 0.0f, 511.0f);
  int y0 = (int)y, x0 = (int)x;
  int y1 = min(y0 + 1, 511), x1 = min(x0 + 1, 511);
  float wy = y - (float)y0, wx = x - (float)x0;
  const float* Mb = M + (size_t)b * RES * RES;
  float v00 = Mb[y0 * RES + x0], v01 = Mb[y0 * RES + x1];
  float v10 = Mb[y1 * RES + x0], v11 = Mb[y1 * RES + x1];
  float top = v00 + (v01 - v00) * wx;
  float bot = v10 + (v11 - v10) * wx;
  out[bk * 256 + t] = top + (bot - top) * wy;
}

// ----------------------------- patch features ------------------------------
__device__ __forceinline__ float pickw(int c, int c0, int c1, float w) {
  float r = 0.0f;
  if (c == c0) r += 1.0f - w;
  if (c == c1) r += w;
  return r;
}

// 2048 blocks x 64 threads. block -> (patch bk = blockIdx>>4, split = &15);
// wave (2/block) owns slab s = split*2 + wave (16 channels).
__global__ void feat_kernel(const float* __restrict__ F,
                            const int* __restrict__ coords,
                            float* __restrict__ out) {
  __shared__ float Wl[2][16][16][16];  // [wave][gy][gx][d16] window, 32 KB
  __shared__ float Tl[2][16][16][16];  // [wave][gy][sx][d16] transpose, 32 KB

  const int bk = blockIdx.x >> 4;      // patch (b*32+k)
  const int split = blockIdx.x & 15;
  const int b = bk >> 5;
  const int lane = threadIdx.x & 31;
  const int wave = threadIdx.x >> 5;   // 2 waves / block
  const int hi = lane >> 4;
  const int m = lane & 15;             // A-matrix row owned by this lane
  const int n = lane & 15;             // B/C/D column owned by this lane
  const int s = split * 2 + wave;      // channel slab 0..31
  const int dbase = s * 16;

  const float ch = (float)coords[bk * 2 + 0];
  const float cw = (float)coords[bk * 2 + 1];
  const float cyn = (ch * 0.125f + 4.0f) * (1.0f / 32.0f) - 1.0f;
  const float cxn = (cw * 0.125f + 4.0f) * (1.0f / 32.0f) - 1.0f;

  // 16-wide window bases (samples span 8 px, so this window covers all taps)
  const float yp0 = clampf((cyn - 0.125f + 1.0f) * 32.0f - 0.5f, 0.0f, 63.0f);
  const float xp0 = clampf((cxn - 0.125f + 1.0f) * 32.0f - 0.5f, 0.0f, 63.0f);
  const int gy_base = min((int)yp0, 48);
  const int gx_base = min((int)xp0, 48);

  const float* Fb = F + (size_t)b * N_ * D_;

  // ---- stage 0: async DMA of the slab window into LDS (ASYNCcnt path) ----
  // 16x16 window rows of 16 channels: 256 x 64B; each lane moves 16B (b128).
  {
    const uint32_t wbase = (uint32_t)(uintptr_t)&Wl[wave][0][0][0];
    const float* corner =
        Fb + ((size_t)gy_base * 64 + gx_base) * D_ + dbase;
    const int q = lane & 3;    // which 4-float quad of the 16-channel row
    const int pl = lane >> 2;  // sub-pair index 0..7
#pragma unroll 1
    for (int i = 0; i < 32; i++) {
      const int p = i * 8 + pl;           // (gy,gx) pair 0..255
      const int gy = p >> 4, gx = p & 15;
      const float* ga = corner + ((size_t)gy * 64 + gx) * D_ + q * 4;
      const uint32_t la = wbase + (uint32_t)(p * 64 + q * 16);
      asm volatile("global_load_async_to_lds_b128 %0, %1, off"
                   :: "v"(la), "v"(ga)
                   : "memory");
    }
    asm volatile("s_wait_asynccnt 0x0" ::: "memory");
  }

  // Interpolation taps for the sample row this lane owns (row m of Wx / Wy).
  const float offm = -0.125f + (float)m * (0.25f / 15.0f);
  float xp = clampf((cxn + offm + 1.0f) * 32.0f - 0.5f, 0.0f, 63.0f);
  int xi0 = (int)xp;
  int xi1 = min(xi0 + 1, 63);
  float wx = xp - (float)xi0;
  const int xc0 = xi0 - gx_base, xc1 = xi1 - gx_base;
  float ypv = clampf((cyn + offm + 1.0f) * 32.0f - 0.5f, 0.0f, 63.0f);
  int yi0 = (int)ypv;
  int yi1 = min(yi0 + 1, 63);
  float wy = ypv - (float)yi0;
  const int yc0 = yi0 - gy_base, yc1 = yi1 - gy_base;

  // A fragments (16x4 f32, ISA 7.12.2): lane m, VGPR v holds K = 4*kk+2*hi+v.
  v2f aX[4], aY[4];
#pragma unroll
  for (int kk = 0; kk < 4; kk++) {
    int c = 4 * kk + 2 * hi;
    aX[kk].x = pickw(c, xc0, xc1, wx);
    aX[kk].y = pickw(c + 1, xc0, xc1, wx);
    aY[kk].x = pickw(c, yc0, yc1, wy);
    aY[kk].y = pickw(c + 1, yc0, yc1, wy);
  }

  // ---- stage 1: T[gy] = Wx * W[gy]  (x-interp), one 16x16x16 per row ----
#pragma unroll 1
  for (int gy = 0; gy < 16; gy++) {
    v8f c8 = {};
#pragma unroll
    for (int kk = 0; kk < 4; kk++) {
      int k0 = 4 * kk + 2 * hi;  // B fragment: K = 4*kk + 2*hi + v, col = n
      v2f bf;
      bf.x = Wl[wave][gy][k0][n];
      bf.y = Wl[wave][gy][k0 + 1][n];
      c8 = __builtin_amdgcn_wmma_f32_16x16x4_f32(false, aX[kk], false, bf,
                                                 (short)0, c8, false, false);
    }
    // D layout: VGPR j -> row j + 8*hi, col n
#pragma unroll
    for (int j = 0; j < 8; j++) Tl[wave][gy][j + 8 * hi][n] = c8[j];
  }
  asm volatile("" ::: "memory");  // keep same-wave DS program order

  // ---- stage 2: out[:, sx, d16] = Wy * T[:, sx, d16]  (y-interp) ----
  float* outp = out + (size_t)bk * 256 * D_ + dbase;
#pragma unroll 1
  for (int sx = 0; sx < 16; sx++) {
    v8f c8 = {};
#pragma unroll
    for (int kk = 0; kk < 4; kk++) {
      int k0 = 4 * kk + 2 * hi;
      v2f bf;
      bf.x = Tl[wave][k0][sx][n];
      bf.y = Tl[wave][k0 + 1][sx][n];
      c8 = __builtin_amdgcn_wmma_f32_16x16x4_f32(false, aY[kk], false, bf,
                                                 (short)0, c8, false, false);
    }
#pragma unroll
    for (int j = 0; j < 8; j++) {
      int sy = j + 8 * hi;
      outp[(size_t)(sy * 16 + sx) * D_ + n] = c8[j];
    }
  }
}

// ------------------------------- launcher ----------------------------------
extern "C" void kernel_launch(void* const* d_in, const int* in_sizes, int n_in,
                              void* d_out, int out_size, void* d_ws,
                              size_t ws_size, hipStream_t stream) {
  const float* features = (const float*)d_in[0];  // [4,4096,512]
  const float* mask     = (const float*)d_in[1];  // [4,1,512,512]
  const float* logits   = (const float*)d_in[2];  // [4,32,1,64,64]
  const int*   coords   = (const int*)d_in[3];    // [4,32,2]

  float* out   = (float*)d_out;
  float* conf  = out;                       // 524288
  float* maskp = out + CONF_N;              // 32768
  float* pfeat = out + CONF_N + MASKP_N;    // 16777216

  conf_kernel<<<CONF_N / 4 / 256, 256, 0, stream>>>(logits, conf);
  mask_kernel<<<B_ * K_, 256, 0, stream>>>(mask, coords, maskp);
  feat_kernel<<<B_ * K_ * 16, 64, 0, stream>>>(features, coords, pfeat);
}